// GAT_25202868093366
// MI455X (gfx1250) — compile-verified
//
#include <hip/hip_runtime.h>
#include <hip/hip_bf16.h>
#include <stdint.h>

typedef float v2f __attribute__((ext_vector_type(2)));
typedef float v8f __attribute__((ext_vector_type(8)));

#define ORD_NEG_INF ((int)0x807FFFFF)   // order-preserving map of -inf
#define SLOPE 0.2f

__device__ __forceinline__ int f_ord(float f) {
    int i = __float_as_int(f);
    return i >= 0 ? i : (i ^ 0x7FFFFFFF);
}
__device__ __forceinline__ float ord_f(int i) {
    return __int_as_float(i >= 0 ? i : (i ^ 0x7FFFFFFF));
}
__device__ __forceinline__ float lrelu(float v) { return v > 0.0f ? v : SLOPE * v; }

// -------------------------------------------------------------------------
// GEMM: C[M,Nc] = A[M,K] @ B[K,Nc] (+ bias), via V_WMMA_F32_16X16X4_F32.
// One wave computes one 16x16 tile; 2 waves (64 threads) per block.
// f32 A frag (16x4): lanes 0-15 rows M, VGPR0/1 = K pair; lanes 16-31 K+2 pair.
// f32 C/D (16x16): VGPR r -> M=r (lanes 0-15) / M=8+r (lanes 16-31), N=lane&15.
// -------------------------------------------------------------------------
__global__ void gemm_wmma_f32(const float* __restrict__ A,
                              const float* __restrict__ B,
                              const float* __restrict__ bias,
                              float* __restrict__ C,
                              int M, int K, int Nc)
{
    const int lane  = threadIdx.x & 31;
    const int wave  = threadIdx.x >> 5;
    const int tileM = (blockIdx.x * 2 + wave) * 16;
    const int tileN = blockIdx.y * 16;
    if (tileM >= M) return;                 // wave-uniform: EXEC stays all-ones
    const int half = lane >> 4;             // 0: K, K+1   1: K+2, K+3
    const int l16  = lane & 15;
    const float* __restrict__ arow = A + (size_t)(tileM + l16) * K;
    const float* __restrict__ bcol = B + tileN + l16;
    v8f acc = {0.f, 0.f, 0.f, 0.f, 0.f, 0.f, 0.f, 0.f};
    for (int k = 0; k < K; k += 4) {
        const int ka = k + half * 2;
        v2f a, b;
        a.x = arow[ka];
        a.y = arow[ka + 1];
        b.x = bcol[(size_t)ka * Nc];
        b.y = bcol[(size_t)(ka + 1) * Nc];
        __builtin_prefetch(arow + ka + 8, 0, 1);   // global_prefetch_b8
        acc = __builtin_amdgcn_wmma_f32_16x16x4_f32(
            /*neg_a=*/false, a, /*neg_b=*/false, b,
            /*c_mod=*/(short)0, acc, /*reuse_a=*/false, /*reuse_b=*/false);
    }
    const float bv = bias ? bias[tileN + l16] : 0.0f;
    const int mbase = tileM + half * 8;
#pragma unroll
    for (int r = 0; r < 8; ++r)
        C[(size_t)(mbase + r) * Nc + tileN + l16] = acc[r] + bv;
}

// -------------------------------------------------------------------------
// Per (node, head) attention dots: a_s[n,h] = <H[n,h,:], att_src[h,:]>, same
// for a_d. One wave per (node,head); float4 per lane + shuffle reduction.
// -------------------------------------------------------------------------
__global__ void attn_dots(const float* __restrict__ H,      // [N, 2, 128]
                          const float* __restrict__ att_s,  // [2, 128]
                          const float* __restrict__ att_d,  // [2, 128]
                          float* __restrict__ a_s, float* __restrict__ a_d,
                          int N)
{
    const int gw   = (int)((blockIdx.x * (size_t)blockDim.x + threadIdx.x) >> 5);
    const int lane = threadIdx.x & 31;
    if (gw >= N * 2) return;                 // wave-uniform
    const int node = gw >> 1, head = gw & 1;
    const float4 hv = *(const float4*)(H + (size_t)node * 256 + head * 128 + lane * 4);
    const float4 ws = *(const float4*)(att_s + head * 128 + lane * 4);
    const float4 wd = *(const float4*)(att_d + head * 128 + lane * 4);
    float s = hv.x * ws.x + hv.y * ws.y + hv.z * ws.z + hv.w * ws.w;
    float d = hv.x * wd.x + hv.y * wd.y + hv.z * wd.z + hv.w * wd.w;
#pragma unroll
    for (int off = 16; off > 0; off >>= 1) {
        s += __shfl_down(s, off, 32);
        d += __shfl_down(d, off, 32);
    }
    if (lane == 0) { a_s[gw] = s; a_d[gw] = d; }
}

// Zero the aggregation buffer and (re)init segment max/sum arrays.
__global__ void init_layer(float* __restrict__ agg, float* __restrict__ seg_sum,
                           int* __restrict__ seg_max, long total, int n2)
{
    const long i = (long)blockIdx.x * blockDim.x + threadIdx.x;
    if (i < total) agg[i] = 0.0f;
    if (i < n2) { seg_sum[i] = 0.0f; seg_max[i] = ORD_NEG_INF; }
}

// Pass 1: segment max of leaky_relu(a_s[src]+a_d[dst]) via int-mapped atomicMax.
__global__ void edge_max(const int* __restrict__ src, const int* __restrict__ dst,
                         int E, int N,
                         const float* __restrict__ a_s, const float* __restrict__ a_d,
                         int* __restrict__ seg_max)
{
    const int e = (int)(blockIdx.x * (size_t)blockDim.x + threadIdx.x);
    const int Et = E + N;
    if (e >= Et) return;
    const int s = (e < E) ? src[e] : (e - E);
    const int d = (e < E) ? dst[e] : (e - E);
#pragma unroll
    for (int h = 0; h < 2; ++h) {
        const float v = lrelu(a_s[s * 2 + h] + a_d[d * 2 + h]);
        atomicMax(&seg_max[d * 2 + h], f_ord(v));
    }
}

// Pass 2: segment sum of exp(e - m[dst]).
__global__ void edge_sum(const int* __restrict__ src, const int* __restrict__ dst,
                         int E, int N,
                         const float* __restrict__ a_s, const float* __restrict__ a_d,
                         const int* __restrict__ seg_max, float* __restrict__ seg_sum)
{
    const int e = (int)(blockIdx.x * (size_t)blockDim.x + threadIdx.x);
    const int Et = E + N;
    if (e >= Et) return;
    const int s = (e < E) ? src[e] : (e - E);
    const int d = (e < E) ? dst[e] : (e - E);
#pragma unroll
    for (int h = 0; h < 2; ++h) {
        const float v = lrelu(a_s[s * 2 + h] + a_d[d * 2 + h]);
        const float m = ord_f(seg_max[d * 2 + h]);
        atomicAdd(&seg_sum[d * 2 + h], __expf(v - m));
    }
}

// Pass 3: out[dst] += alpha * h[src]; one wave per edge, 8 channels per lane.
__global__ void edge_scatter(const int* __restrict__ src, const int* __restrict__ dst,
                             int E, int N,
                             const float* __restrict__ a_s, const float* __restrict__ a_d,
                             const int* __restrict__ seg_max, const float* __restrict__ seg_sum,
                             const float* __restrict__ H, float* __restrict__ out)
{
    const int gw   = (int)((blockIdx.x * (size_t)blockDim.x + threadIdx.x) >> 5);
    const int lane = threadIdx.x & 31;
    const int Et = E + N;
    if (gw >= Et) return;                    // wave-uniform
    const int s = (gw < E) ? src[gw] : (gw - E);
    const int d = (gw < E) ? dst[gw] : (gw - E);
    const int head = lane >> 4;              // lanes 0-15 -> head 0, 16-31 -> head 1
    const float v = lrelu(a_s[s * 2 + head] + a_d[d * 2 + head]);
    const float m = ord_f(seg_max[d * 2 + head]);
    const float alpha = __expf(v - m) / (seg_sum[d * 2 + head] + 1e-16f);
    const float* __restrict__ hs = H + (size_t)s * 256 + lane * 8;
    float* __restrict__ od = out + (size_t)d * 256 + lane * 8;
    const float4 h0 = *(const float4*)(hs);
    const float4 h1 = *(const float4*)(hs + 4);
    atomicAdd(od + 0, alpha * h0.x);
    atomicAdd(od + 1, alpha * h0.y);
    atomicAdd(od + 2, alpha * h0.z);
    atomicAdd(od + 3, alpha * h0.w);
    atomicAdd(od + 4, alpha * h1.x);
    atomicAdd(od + 5, alpha * h1.y);
    atomicAdd(od + 6, alpha * h1.z);
    atomicAdd(od + 7, alpha * h1.w);
}

// out = relu(in + bias[col]); col = i & 255 (256 channels).
__global__ void bias_relu(const float* __restrict__ in, const float* __restrict__ bias,
                          float* __restrict__ out, long total)
{
    const long i = (long)blockIdx.x * blockDim.x + threadIdx.x;
    if (i >= total) return;
    const float v = in[i] + bias[i & 255];
    out[i] = v > 0.0f ? v : 0.0f;
}

// Final: sigmoid(C1[n,:] . Wc2 + bc2); one wave per node.
__global__ void final_head(const float* __restrict__ C1, const float* __restrict__ Wc2,
                           const float* __restrict__ bc2, float* __restrict__ out, int N)
{
    const int gw   = (int)((blockIdx.x * (size_t)blockDim.x + threadIdx.x) >> 5);
    const int lane = threadIdx.x & 31;
    if (gw >= N) return;                     // wave-uniform
    const float4 c = *(const float4*)(C1 + (size_t)gw * 128 + lane * 4);
    const float4 w = *(const float4*)(Wc2 + lane * 4);
    float s = c.x * w.x + c.y * w.y + c.z * w.z + c.w * w.w;
#pragma unroll
    for (int off = 16; off > 0; off >>= 1) s += __shfl_down(s, off, 32);
    if (lane == 0) out[gw] = 1.0f / (1.0f + __expf(-(s + bc2[0])));
}

// -------------------------------------------------------------------------
extern "C" void kernel_launch(void* const* d_in, const int* in_sizes, int n_in,
                              void* d_out, int out_size, void* d_ws, size_t ws_size,
                              hipStream_t stream)
{
    const float* x    = (const float*)d_in[0];
    const int*   ei   = (const int*)d_in[1];
    const float* W1   = (const float*)d_in[2];
    const float* as1  = (const float*)d_in[3];
    const float* ad1  = (const float*)d_in[4];
    const float* b1   = (const float*)d_in[5];
    const float* W2   = (const float*)d_in[6];
    const float* as2  = (const float*)d_in[7];
    const float* ad2  = (const float*)d_in[8];
    const float* b2   = (const float*)d_in[9];
    const float* Wc1  = (const float*)d_in[10];
    const float* bc1  = (const float*)d_in[11];
    const float* Wc2  = (const float*)d_in[12];
    const float* bc2  = (const float*)d_in[13];
    float* outp = (float*)d_out;

    const int N = in_sizes[0] / 128;     // 100000
    const int E = in_sizes[1] / 2;       // 1600000
    const int Et = E + N;
    const int* src = ei;
    const int* dst = ei + E;

    auto au = [](size_t v) { return (v + 255) & ~(size_t)255; };
    char* p = (char*)d_ws;
    float* bufA   = (float*)p; p += au((size_t)N * 256 * sizeof(float));
    float* bufB   = (float*)p; p += au((size_t)N * 256 * sizeof(float));
    float* bufC1  = (float*)p; p += au((size_t)N * 128 * sizeof(float));
    float* aS     = (float*)p; p += au((size_t)N * 2 * sizeof(float));
    float* aD     = (float*)p; p += au((size_t)N * 2 * sizeof(float));
    float* segSum = (float*)p; p += au((size_t)N * 2 * sizeof(float));
    int*   segMax = (int*)p;   p += au((size_t)N * 2 * sizeof(int));

    const long totNC = (long)N * 256;
    const int Mtiles = (N + 15) / 16;                       // 6250, exact
    const dim3 gemmBlk(64);                                 // 2 waves
    const dim3 g256((Mtiles + 1) / 2, 256 / 16);
    const dim3 g128((Mtiles + 1) / 2, 128 / 16);
    const int attnBlocks = (int)(((size_t)2 * N * 32 + 255) / 256);
    const int initBlocks = (int)((totNC + 255) / 256);
    const int edgeBlocks = (Et + 255) / 256;
    const int scatBlocks = (int)(((size_t)Et * 32 + 255) / 256);
    const int headBlocks = (int)(((size_t)N * 32 + 255) / 256);

    // ---------------- Layer 1 ----------------
    gemm_wmma_f32<<<g256, gemmBlk, 0, stream>>>(x, W1, nullptr, bufA, N, 128, 256);
    attn_dots<<<attnBlocks, 256, 0, stream>>>(bufA, as1, ad1, aS, aD, N);
    init_layer<<<initBlocks, 256, 0, stream>>>(bufB, segSum, segMax, totNC, N * 2);
    edge_max<<<edgeBlocks, 256, 0, stream>>>(src, dst, E, N, aS, aD, segMax);
    edge_sum<<<edgeBlocks, 256, 0, stream>>>(src, dst, E, N, aS, aD, segMax, segSum);
    edge_scatter<<<scatBlocks, 256, 0, stream>>>(src, dst, E, N, aS, aD, segMax, segSum,
                                                 bufA, bufB);
    bias_relu<<<initBlocks, 256, 0, stream>>>(bufB, b1, bufA, totNC);

    // ---------------- Layer 2 ----------------
    gemm_wmma_f32<<<g256, gemmBlk, 0, stream>>>(bufA, W2, nullptr, bufB, N, 256, 256);
    attn_dots<<<attnBlocks, 256, 0, stream>>>(bufB, as2, ad2, aS, aD, N);
    init_layer<<<initBlocks, 256, 0, stream>>>(bufA, segSum, segMax, totNC, N * 2);
    edge_max<<<edgeBlocks, 256, 0, stream>>>(src, dst, E, N, aS, aD, segMax);
    edge_sum<<<edgeBlocks, 256, 0, stream>>>(src, dst, E, N, aS, aD, segMax, segSum);
    edge_scatter<<<scatBlocks, 256, 0, stream>>>(src, dst, E, N, aS, aD, segMax, segSum,
                                                 bufB, bufA);
    bias_relu<<<initBlocks, 256, 0, stream>>>(bufA, b2, bufB, totNC);

    // ---------------- Classifier ----------------
    gemm_wmma_f32<<<g128, gemmBlk, 0, stream>>>(bufB, Wc1, bc1, bufC1, N, 256, 128);
    final_head<<<headBlocks, 256, 0, stream>>>(bufC1, Wc2, bc2, outp, N);
}